// MultiHeadTEAttention_60919816126953
// MI455X (gfx1250) — compile-verified
//
#include <hip/hip_runtime.h>

#define NB 2
#define NQL 1024
#define NKL 1024
#define DX 256
#define NH 8
#define DD 32
#define KHW 32
#define DTW 2
#define SCALE_F 0.17677669529663687f

typedef float v2f __attribute__((ext_vector_type(2)));
typedef float v8f __attribute__((ext_vector_type(8)));

__device__ __forceinline__ v8f wmma4(v2f a, v2f b, v8f c) {
  // D = A(16x4 f32) * B(4x16 f32) + C(16x16 f32)
  return __builtin_amdgcn_wmma_f32_16x16x4_f32(false, a, false, b, (short)0, c,
                                               false, false);
}

__device__ __forceinline__ float xmax16(float x) {
#pragma unroll
  for (int o = 1; o < 16; o <<= 1) x = fmaxf(x, __shfl_xor(x, o, 32));
  return x;
}

// ---------------------------------------------------------------------------
// Kernel 1: Q/K/V projections, stored as [m, h, n, 32]
// grid (2048/32, 256/64, 3), block 256 (8 waves, each a 16x16 out tile)
// ---------------------------------------------------------------------------
__global__ void __launch_bounds__(256) proj_qkv(
    const float* __restrict__ xq, const float* __restrict__ xk,
    const float* __restrict__ xv, const float* __restrict__ Wq,
    const float* __restrict__ Wk, const float* __restrict__ Wv,
    float* __restrict__ Qh, float* __restrict__ Kh, float* __restrict__ Vh) {
  const float* A;
  const float* W;
  float* dst;
  if (blockIdx.z == 0) {
    A = xq; W = Wq; dst = Qh;
  } else if (blockIdx.z == 1) {
    A = xk; W = Wk; dst = Kh;
  } else {
    A = xv; W = Wv; dst = Vh;
  }
  const int w = threadIdx.x >> 5;
  const int lane = threadIdx.x & 31;
  const int hl = lane >> 4, l16 = lane & 15;
  const int r0 = (blockIdx.x * 2 + (w >> 2)) * 16;
  const int c0 = (blockIdx.y * 4 + (w & 3)) * 16;

  v8f acc = {};
  const float* arow = A + (size_t)(r0 + l16) * DX;
#pragma unroll 4
  for (int kc = 0; kc < DX / 4; ++kc) {
    const int kidx = kc * 4 + hl * 2;
    v2f a = *(const v2f*)(arow + kidx);
    v2f b;
    b.x = W[(size_t)kidx * DX + c0 + l16];
    b.y = W[(size_t)(kidx + 1) * DX + c0 + l16];
    acc = wmma4(a, b, acc);
  }
#pragma unroll
  for (int j = 0; j < 8; ++j) {
    const int r = r0 + j + hl * 8;          // row (m*1024+q)
    const int c = c0 + l16;                 // col (h*32+d)
    const int mb = r >> 10, q = r & 1023;
    const int h = c >> 5, d = c & 31;
    dst[(((size_t)mb * NH + h) * NQL + q) * DD + d] = acc[j];
  }
}

// ---------------------------------------------------------------------------
// Kernel 2: fused TE-bias + flash attention + phi accumulation
// grid (NQ/16, M), block 256 = 8 waves (wave h handles head h).
// Softmax statistics (l, sum p*tkx, sum p*tky) are produced by an extra
// WMMA against a virtual V-extension [1 | tkx | tky] instead of shuffle
// reductions; only the row-max uses cross-lane ops.
// ---------------------------------------------------------------------------
__global__ void __launch_bounds__(256) attn_fused(
    const float* __restrict__ Qh, const float* __restrict__ Kh,
    const float* __restrict__ Vh, const float* __restrict__ tq,
    const float* __restrict__ tk, const float* __restrict__ Wk1,
    const float* __restrict__ bk1, const float* __restrict__ Wk2,
    const float* __restrict__ bk2, const float* __restrict__ Wphi,
    float* __restrict__ Oheads, float* __restrict__ Aphi) {
  __shared__ float sWk1[DTW * KHW];
  __shared__ float sbk1[KHW];
  __shared__ float sWk2[KHW * NH];
  __shared__ float sbk2[NH];
  __shared__ float kd[NH][16][18];  // padded: halves hit disjoint banks
  __shared__ float pb[NH][16][17];  // padded: transposed reads conflict-free

  const int tid = threadIdx.x;
  const int h = tid >> 5;
  const int lane = tid & 31;
  const int hl = lane >> 4, l16 = lane & 15;
  const int q0 = blockIdx.x * 16;
  const int mb = blockIdx.y;

  if (tid < DTW * KHW) sWk1[tid] = Wk1[tid];
  if (tid < KHW) sbk1[tid] = bk1[tid];
  if (tid < KHW * NH) sWk2[tid] = Wk2[tid];
  if (tid < NH) sbk2[tid] = bk2[tid];

  // cooperative pair-MLP mapping: thread -> (query qi, key ki) of the tile
  const int qi = tid & 15, ki = tid >> 4;
  const float tqx = tq[((size_t)mb * NQL + q0 + qi) * DTW + 0];
  const float tqy = tq[((size_t)mb * NQL + q0 + qi) * DTW + 1];

  const float* qbase = Qh + (((size_t)mb * NH + h) * NQL) * DD;
  const float* kbase = Kh + (((size_t)mb * NH + h) * NKL) * DD;
  const float* vbase = Vh + (((size_t)mb * NH + h) * NKL) * DD;
  const float* tkb = tk + (size_t)mb * NKL * DTW;

  // Q tile resident as A-fragments (16x32 -> 8 chunks of 16x4)
  v2f aq[8];
#pragma unroll
  for (int c = 0; c < 8; ++c)
    aq[c] = *(const v2f*)(qbase + (size_t)(q0 + l16) * DD + c * 4 + hl * 2);

  v8f olo = {}, ohi = {};
  v8f ostat = {};  // col0 = l = sum p, col1 = sum p*tkx, col2 = sum p*tky
  float mrow[8];
#pragma unroll
  for (int j = 0; j < 8; ++j) mrow[j] = -1e30f;

  for (int kt = 0; kt < NKL / 16; ++kt) {
    const int k0 = kt * 16;
    __syncthreads();  // previous kd consumers done (also covers weight stage)
    {                 // kernel-MLP bias, computed once per pair, all heads
      const float tkx = tkb[(size_t)(k0 + ki) * DTW + 0];
      const float tky = tkb[(size_t)(k0 + ki) * DTW + 1];
      const float dx = tqx - tkx, dy = tqy - tky;
      float outh[NH];
#pragma unroll
      for (int hh = 0; hh < NH; ++hh) outh[hh] = sbk2[hh];
#pragma unroll 4
      for (int u = 0; u < KHW; ++u) {
        float hu = fmaf(dx, sWk1[u], fmaf(dy, sWk1[KHW + u], sbk1[u]));
        hu = fmaxf(hu, 0.f);
#pragma unroll
        for (int hh = 0; hh < NH; ++hh)
          outh[hh] = fmaf(hu, sWk2[u * NH + hh], outh[hh]);
      }
#pragma unroll
      for (int hh = 0; hh < NH; ++hh) kd[hh][qi][ki] = outh[hh];
    }
    __syncthreads();

    // token dots: Q(16x32) @ K^T(32x16)
    v8f acc = {};
#pragma unroll
    for (int c = 0; c < 8; ++c) {
      v2f b = *(const v2f*)(kbase + (size_t)(k0 + l16) * DD + c * 4 + hl * 2);
      acc = wmma4(aq[c], b, acc);
    }

    // online softmax: only the row-max needs cross-lane reduction
#pragma unroll
    for (int j = 0; j < 8; ++j) {
      const int qrow = j + hl * 8;  // D-layout row for this half-wave
      float x = fmaf(acc[j], SCALE_F, kd[h][qrow][l16]);
      const float rm = xmax16(x);
      const float mnew = fmaxf(mrow[j], rm);
      const float p = __expf(x - mnew);
      const float alpha = __expf(mrow[j] - mnew);
      mrow[j] = mnew;
      olo[j] *= alpha;
      ohi[j] *= alpha;
      ostat[j] *= alpha;
      pb[h][qrow][l16] = p;  // wave-private staging, D-layout -> A-layout
    }

    // P(16x16) @ [ V(16x32) | 1 | tkx | tky ] : 4 K-chunks x 3 output tiles
#pragma unroll
    for (int c = 0; c < 4; ++c) {
      const int kk2 = c * 4 + hl * 2;
      v2f ap;
      ap.x = pb[h][l16][kk2];
      ap.y = pb[h][l16][kk2 + 1];
      v2f bl, bh;
      bl.x = vbase[(size_t)(k0 + kk2) * DD + l16];
      bl.y = vbase[(size_t)(k0 + kk2 + 1) * DD + l16];
      bh.x = vbase[(size_t)(k0 + kk2) * DD + 16 + l16];
      bh.y = vbase[(size_t)(k0 + kk2 + 1) * DD + 16 + l16];
      // virtual columns: [1 | tkx | tky | 0...] for the two keys of this lane
      v2f tkp0 = *(const v2f*)(tkb + (size_t)(k0 + kk2) * DTW);
      v2f tkp1 = *(const v2f*)(tkb + (size_t)(k0 + kk2 + 1) * DTW);
      v2f bs;
      bs.x = (l16 == 0) ? 1.0f : (l16 == 1) ? tkp0.x : (l16 == 2) ? tkp0.y : 0.0f;
      bs.y = (l16 == 0) ? 1.0f : (l16 == 1) ? tkp1.x : (l16 == 2) ? tkp1.y : 0.0f;
      olo = wmma4(ap, bl, olo);
      ohi = wmma4(ap, bh, ohi);
      ostat = wmma4(ap, bs, ostat);
    }
  }

  // epilogue: broadcast l/T from stat-tile columns 0/1/2, normalize, store
  const int cbase = lane & 16;  // column-0 lane of this half
#pragma unroll
  for (int j = 0; j < 8; ++j) {
    const int q = q0 + j + hl * 8;
    const float lj = __shfl(ostat[j], cbase + 0, 32);
    const float t0 = __shfl(ostat[j], cbase + 1, 32);
    const float t1 = __shfl(ostat[j], cbase + 2, 32);
    const float invl = 1.f / lj;
    const size_t ob = (((size_t)mb * NH + h) * NQL + q) * DD;
    Oheads[ob + l16] = olo[j] * invl;
    Oheads[ob + 16 + l16] = ohi[j] * invl;
    if (l16 == 0) {  // sum_k attn*diff = tq - T/l  (per dt)
      const float qx = tq[((size_t)mb * NQL + q) * DTW + 0];
      const float qy = tq[((size_t)mb * NQL + q) * DTW + 1];
      const float ax = qx - t0 * invl;
      const float ay = qy - t1 * invl;
      const size_t pi = (((size_t)mb * NH + h) * NQL + q) * DTW;
      Aphi[pi + 0] = Wphi[h * DTW + 0] * ax;
      Aphi[pi + 1] = Wphi[h * DTW + 1] * ay;
    }
  }
}

// ---------------------------------------------------------------------------
// Kernel 3: out = concat_heads(O) @ Wo + bo      grid (64,4), block 256
// ---------------------------------------------------------------------------
__global__ void __launch_bounds__(256) out_proj(
    const float* __restrict__ Oheads, const float* __restrict__ Wo,
    const float* __restrict__ bo, float* __restrict__ out) {
  const int w = threadIdx.x >> 5;
  const int lane = threadIdx.x & 31;
  const int hl = lane >> 4, l16 = lane & 15;
  const int r0 = (blockIdx.x * 2 + (w >> 2)) * 16;
  const int c0 = (blockIdx.y * 4 + (w & 3)) * 16;
  const int r = r0 + l16;
  const int mb = r >> 10, q = r & 1023;

  v8f acc = {};
#pragma unroll 4
  for (int kc = 0; kc < DX / 4; ++kc) {
    const int kidx = kc * 4 + hl * 2;  // even -> both elems in same head
    const int hh = kidx >> 5, d = kidx & 31;
    v2f a = *(const v2f*)(Oheads + (((size_t)mb * NH + hh) * NQL + q) * DD + d);
    v2f b;
    b.x = Wo[(size_t)kidx * DX + c0 + l16];
    b.y = Wo[(size_t)(kidx + 1) * DX + c0 + l16];
    acc = wmma4(a, b, acc);
  }
  const float bias = bo[c0 + l16];
#pragma unroll
  for (int j = 0; j < 8; ++j) {
    const int rr = r0 + j + hl * 8;
    out[(size_t)rr * DX + c0 + l16] = acc[j] + bias;
  }
}

// ---------------------------------------------------------------------------
// Kernel 4: tq_new = tq + (1/nk)*( sum_h Wphi[h]*A_h + bphi*(nk*tq - sum_k tk) )
// ---------------------------------------------------------------------------
__global__ void __launch_bounds__(256) tq_update(
    const float* __restrict__ tq, const float* __restrict__ tk,
    const float* __restrict__ Aphi, const float* __restrict__ bphi,
    float* __restrict__ tq_out) {
  __shared__ float red[256];
  __shared__ float sum0s, sum1s;
  const int mb = blockIdx.x;
  const int tid = threadIdx.x;
  float s0 = 0.f, s1 = 0.f;
  for (int k = tid; k < NKL; k += 256) {
    s0 += tk[((size_t)mb * NKL + k) * DTW + 0];
    s1 += tk[((size_t)mb * NKL + k) * DTW + 1];
  }
  red[tid] = s0;
  __syncthreads();
  for (int o = 128; o > 0; o >>= 1) {
    if (tid < o) red[tid] += red[tid + o];
    __syncthreads();
  }
  if (tid == 0) sum0s = red[0];
  __syncthreads();
  red[tid] = s1;
  __syncthreads();
  for (int o = 128; o > 0; o >>= 1) {
    if (tid < o) red[tid] += red[tid + o];
    __syncthreads();
  }
  if (tid == 0) sum1s = red[0];
  __syncthreads();
  const float sum0 = sum0s, sum1 = sum1s;
  const float b0 = bphi[0], b1 = bphi[1];
  const float inv = 1.0f / (float)NKL;
  for (int q = tid; q < NQL; q += 256) {
    float a0 = 0.f, a1 = 0.f;
#pragma unroll
    for (int hh = 0; hh < NH; ++hh) {
      const size_t idx = (((size_t)mb * NH + hh) * NQL + q) * DTW;
      a0 += Aphi[idx + 0];
      a1 += Aphi[idx + 1];
    }
    const float qx = tq[((size_t)mb * NQL + q) * DTW + 0];
    const float qy = tq[((size_t)mb * NQL + q) * DTW + 1];
    tq_out[((size_t)mb * NQL + q) * DTW + 0] =
        qx + inv * (a0 + b0 * ((float)NKL * qx - sum0));
    tq_out[((size_t)mb * NQL + q) * DTW + 1] =
        qy + inv * (a1 + b1 * ((float)NKL * qy - sum1));
  }
}

// ---------------------------------------------------------------------------
extern "C" void kernel_launch(void* const* d_in, const int* in_sizes, int n_in,
                              void* d_out, int out_size, void* d_ws,
                              size_t ws_size, hipStream_t stream) {
  const float* xq = (const float*)d_in[0];
  const float* xk = (const float*)d_in[1];
  const float* xv = (const float*)d_in[2];
  const float* tq = (const float*)d_in[3];
  const float* tk = (const float*)d_in[4];
  const float* Wq = (const float*)d_in[5];
  const float* Wk = (const float*)d_in[6];
  const float* Wv = (const float*)d_in[7];
  const float* Wo = (const float*)d_in[8];
  const float* bo = (const float*)d_in[9];
  const float* Wk1 = (const float*)d_in[10];
  const float* bk1 = (const float*)d_in[11];
  const float* Wk2 = (const float*)d_in[12];
  const float* bk2 = (const float*)d_in[13];
  const float* Wphi = (const float*)d_in[14];
  const float* bphi = (const float*)d_in[15];

  float* ws = (float*)d_ws;
  const size_t nqh = (size_t)NB * NH * NQL * DD;  // 524288 floats
  float* Qh = ws;
  float* Kh = Qh + nqh;
  float* Vh = Kh + nqh;
  float* Oh = Vh + nqh;
  float* Aphi = Oh + nqh;  // NB*NH*NQL*DTW = 32768 floats

  float* out = (float*)d_out;
  float* tqo = out + (size_t)NB * NQL * DX;

  proj_qkv<<<dim3(64, 4, 3), 256, 0, stream>>>(xq, xk, xv, Wq, Wk, Wv, Qh, Kh,
                                               Vh);
  attn_fused<<<dim3(NQL / 16, NB), 256, 0, stream>>>(
      Qh, Kh, Vh, tq, tk, Wk1, bk1, Wk2, bk2, Wphi, Oh, Aphi);
  out_proj<<<dim3(64, 4), 256, 0, stream>>>(Oh, Wo, bo, out);
  tq_update<<<dim3(NB), 256, 0, stream>>>(tq, tk, Aphi, bphi, tqo);
}